// EncoderRelGraphConvHomo_25890062860622
// MI455X (gfx1250) — compile-verified
//
#include <hip/hip_runtime.h>

typedef float v2f __attribute__((ext_vector_type(2)));
typedef float v8f __attribute__((ext_vector_type(8)));

#define HDIM   128
#define NRELS  8
#define NBASES 4
#define APAD   132   // LDS row stride (floats): 128 + 4 pad -> conflict-free b64 reads

// ---------------------------------------------------------------------------
// Pre-swizzle W[r] = sum_b comp[r,b]*basis[b] into WMMA B-fragment order.
// Layout: wsw[r][ctile][s][lane][2]  (r*16384 + ((ctile*32+s)*32+lane)*2)
// For V_WMMA_F32_16X16X4_F32, B is 4(K)x16(N):
//   lanes 0-15  : N=lane,    V0 = K=4s+0, V1 = K=4s+1
//   lanes 16-31 : N=lane-16, V0 = K=4s+2, V1 = K=4s+3
// ---------------------------------------------------------------------------
__global__ void prepack_w(const float* __restrict__ basis,
                          const float* __restrict__ comp,
                          float* __restrict__ wsw) {
  int f = blockIdx.x * blockDim.x + threadIdx.x;   // 65536 threads total
  int lane  = f & 31;
  int s     = (f >> 5) & 31;
  int ctile = (f >> 10) & 7;
  int r     = f >> 13;
  if (r >= NRELS) return;
  int col = ctile * 16 + (lane & 15);
  int k   = s * 4 + ((lane & 16) ? 2 : 0);
  float w0 = 0.0f, w1 = 0.0f;
  for (int b = 0; b < NBASES; ++b) {
    float c = comp[r * NBASES + b];
    w0 += c * basis[((size_t)b * HDIM + k) * HDIM + col];
    w1 += c * basis[((size_t)b * HDIM + k + 1) * HDIM + col];
  }
  float* p = wsw + (size_t)r * 16384 + ((size_t)(ctile * 32 + s) * 32 + lane) * 2;
  p[0] = w0;
  p[1] = w1;
}

// ---------------------------------------------------------------------------
// Hr = h @ W[r] with V_WMMA_F32_16X16X4_F32 (exact fp32).
// Block = 256 threads = 8 waves; block owns one 16-row strip, wave w owns
// column tile w. The 16x128 A strip is staged once into LDS (8 KB, padded
// stride 132) by all 256 threads, then each wave reads its A fragments as
// conflict-free ds_load_b64. No EXEC divergence in the K-loop.
// ---------------------------------------------------------------------------
__global__ void gemm_wmma_f32(const float* __restrict__ hmat,
                              const float* __restrict__ wsw_r,
                              float* __restrict__ out, int nrows) {
  __shared__ float atile[16 * APAD];

  const int tid  = threadIdx.x;
  const int row0 = blockIdx.x * 16;

  // ---- cooperative stage: each thread moves 8 contiguous floats ----
  {
    int idx  = tid * 8;            // 0..2040
    int r    = idx >> 7;           // tile row 0..15
    int c    = idx & 127;          // col 0..120 (multiple of 8)
    int grow = row0 + r;
    if (grow >= nrows) grow = nrows - 1;   // clamp: dup load, never stored
    const float4* g = (const float4*)(hmat + (size_t)grow * HDIM + c);
    float4 v0 = g[0];
    float4 v1 = g[1];
    float* l = atile + r * APAD + c;       // 8-byte aligned (APAD*4 % 16 == 0? use f4)
    *(float4*)(l + 0) = v0;
    *(float4*)(l + 4) = v1;
  }
  __syncthreads();

  const int  lane  = tid & 31;
  const int  ctile = tid >> 5;     // 0..7
  const bool hi    = lane >= 16;
  const int  col   = ctile * 16 + (lane & 15);
  const float* arow_lds = atile + (lane & 15) * APAD + (hi ? 2 : 0);
  const float* bbase    = wsw_r + (size_t)ctile * 2048;   // 32 steps * 64 floats

  v8f acc = {};
#pragma unroll
  for (int s = 0; s < 32; ++s) {
    v2f a = *(const v2f*)(arow_lds + 4 * s);            // ds_load_b64
    v2f b = *(const v2f*)(bbase + s * 64 + lane * 2);   // global_load_b64 (L2 hit)
    acc = __builtin_amdgcn_wmma_f32_16x16x4_f32(
        /*neg_a=*/false, a, /*neg_b=*/false, b,
        /*c_mod=*/(short)0, acc, /*reuse_a=*/false, /*reuse_b=*/false);
  }

  // D layout: lanes 0-15 rows M=0..7, lanes 16-31 rows M=8..15, N=col
  int mbase = row0 + (hi ? 8 : 0);
#pragma unroll
  for (int j = 0; j < 8; ++j) {
    int row = mbase + j;
    if (row < nrows) out[(size_t)row * HDIM + col] = acc[j];
  }
}

// ---------------------------------------------------------------------------
// For edges of relation `rel`: agg[dst] += Hr[src] * norm  (atomic f32 add).
// One wave per edge, float4 per lane (32 lanes * 4 = 128).
// ---------------------------------------------------------------------------
__global__ void scatter_edges(const float* __restrict__ hr,
                              const int* __restrict__ srcv,
                              const int* __restrict__ dstv,
                              const int* __restrict__ etyp,
                              const float* __restrict__ norm,
                              float* __restrict__ agg,
                              int rel, int nedges) {
  int gid  = blockIdx.x * blockDim.x + threadIdx.x;
  int e    = gid >> 5;
  int lane = gid & 31;
  if (e >= nedges) return;
  if (etyp[e] != rel) return;
  int s = srcv[e];
  int d = dstv[e];
  float nr = norm[e];
  const float4 v = *(const float4*)(hr + (size_t)s * HDIM + lane * 4);
  float* p = agg + (size_t)d * HDIM + lane * 4;
  atomicAdd(p + 0, v.x * nr);
  atomicAdd(p + 1, v.y * nr);
  atomicAdd(p + 2, v.z * nr);
  atomicAdd(p + 3, v.w * nr);
}

__global__ void fill_zero(float* __restrict__ p, size_t n) {
  size_t i = (size_t)blockIdx.x * blockDim.x + threadIdx.x;
  if (i < n) p[i] = 0.0f;
}

__global__ void bias_relu(float* __restrict__ p, const float* __restrict__ bias,
                          size_t n) {
  size_t i = (size_t)blockIdx.x * blockDim.x + threadIdx.x;
  if (i < n) {
    float x = p[i] + bias[i & (HDIM - 1)];
    p[i] = x > 0.0f ? x : 0.0f;
  }
}

// ---------------------------------------------------------------------------
extern "C" void kernel_launch(void* const* d_in, const int* in_sizes, int n_in,
                              void* d_out, int out_size, void* d_ws, size_t ws_size,
                              hipStream_t stream) {
  const float* feats  = (const float*)d_in[0];
  const int*   src    = (const int*)d_in[1];
  const int*   dstv   = (const int*)d_in[2];
  const int*   etyp   = (const int*)d_in[3];
  const float* norm   = (const float*)d_in[4];
  const float* basis0 = (const float*)d_in[5];
  const float* comp0  = (const float*)d_in[6];
  const float* bias0  = (const float*)d_in[7];
  const float* basis1 = (const float*)d_in[8];
  const float* comp1  = (const float*)d_in[9];
  const float* bias1  = (const float*)d_in[10];
  float* out = (float*)d_out;

  const int N = in_sizes[0] / HDIM;
  const int E = in_sizes[1];

  float* wsw = (float*)d_ws;                       // 131072 floats (512 KB)
  float* hrb = wsw + (size_t)NRELS * 16384;        // N*128 floats (25.6 MB)
  float* h1  = hrb + (size_t)N * HDIM;             // N*128 floats (25.6 MB)

  const size_t nfeat = (size_t)N * HDIM;
  const int fill_blocks = (int)((nfeat + 255) / 256);
  const int ntiles = (N + 15) / 16;
  const int eblocks = (int)(((size_t)E * 32 + 255) / 256);

  // ---- Layer 0: feats -> h1 ----
  prepack_w<<<256, 256, 0, stream>>>(basis0, comp0, wsw);
  fill_zero<<<fill_blocks, 256, 0, stream>>>(h1, nfeat);
  for (int r = 0; r < NRELS; ++r) {
    gemm_wmma_f32<<<ntiles, 256, 0, stream>>>(feats, wsw + (size_t)r * 16384, hrb, N);
    scatter_edges<<<eblocks, 256, 0, stream>>>(hrb, src, dstv, etyp, norm, h1, r, E);
  }
  bias_relu<<<fill_blocks, 256, 0, stream>>>(h1, bias0, nfeat);

  // ---- Layer 1: h1 -> out ----
  prepack_w<<<256, 256, 0, stream>>>(basis1, comp1, wsw);
  fill_zero<<<fill_blocks, 256, 0, stream>>>(out, nfeat);
  for (int r = 0; r < NRELS; ++r) {
    gemm_wmma_f32<<<ntiles, 256, 0, stream>>>(h1, wsw + (size_t)r * 16384, hrb, N);
    scatter_edges<<<eblocks, 256, 0, stream>>>(hrb, src, dstv, etyp, norm, out, r, E);
  }
  bias_relu<<<fill_blocks, 256, 0, stream>>>(out, bias1, nfeat);
}